// ConvSPE_47330539602331
// MI455X (gfx1250) — compile-verified
//
#include <hip/hip_runtime.h>

// ConvSPE for MI455X (gfx1250, wave32).
// Strategy: depthwise 200-tap FIR recast as banded-Toeplitz GEMM on
// V_WMMA_F32_16X16X4_F32 (fp32-exact, 93% band density), fused with the
// d-reduction against queries (qhat) and the sliced d-reduction against
// keys (khat), single pass over `noise` per l-chunk.

typedef __attribute__((ext_vector_type(2))) float v2f;
typedef __attribute__((ext_vector_type(8))) float v8f;

#define B_    4
#define D_    64
#define L_    4096
#define K_    200
#define NUM_  32
#define TN_   (L_ + 2 * K_)        // 4496 noise row length
#define CHUNK 128                  // l per workgroup
#define CONTR 216                  // Toeplitz contraction (K + 16), 54 * 4
#define TILE_T 344                 // CHUNK + CONTR - 16 - 8 pad... = staged t extent
#define PITCH 348                  // LDS row pitch (16B aligned, bank-conflict free)
#define WL_LEN 232                 // 16 zero | 200 weights | 16 zero
#define NWAVE 8
#define BLOCK (NWAVE * 32)

__global__ __launch_bounds__(BLOCK) void convspe_wmma_kernel(
    const float* __restrict__ qry, const float* __restrict__ key,
    const float* __restrict__ noise, const float* __restrict__ cw,
    float* __restrict__ outq, float* __restrict__ outk) {
  __shared__ float Xs[16 * PITCH];  // 16 bn rows x t-slab for current d
  __shared__ float wl[WL_LEN];      // zero-padded weights for channel d

  const int tid   = threadIdx.x;
  const int lane  = tid & 31;
  const int wave  = tid >> 5;
  const int chunk = blockIdx.x;          // 0..31 : l-chunk
  const int bg    = blockIdx.y;          // 0..7  : bn-group of 16
  const int l0c   = chunk * CHUNK;       // chunk base (abs l, == abs t origin)
  const int bn0   = bg * 16;
  const int b     = bn0 / NUM_;          // 16 | 32, so group maps to one batch
  const int col   = lane & 15;           // N index (l offset) / A row (m)
  const int hi    = lane >> 4;
  const int kbase = hi * 2;              // f32 wmma: upper half-lanes hold k=2,3
  const int lrel  = wave * 16;           // this wave's l-tile within chunk
  const int l_abs = l0c + lrel + col;

  const float inv_scale = 0.022097086912079612f;  // 1/sqrt(NUM*D) = 1/sqrt(2048)

  float qacc[8], kacc[8];
#pragma unroll
  for (int v = 0; v < 8; ++v) { qacc[v] = 0.f; kacc[v] = 0.f; }

  for (int d = 0; d < D_; ++d) {
    __syncthreads();  // protect previous iteration's readers

    // ---- stage zero-padded channel weights: wl[16 + j] = cw[d*K + j] ----
    if (tid < WL_LEN) {
      int j = tid - 16;
      wl[tid] = (j >= 0 && j < K_) ? cw[d * K_ + j] : 0.0f;
    }
    // ---- stage noise slab: 16 rows x TILE_T floats (float4, coalesced) ----
    {
      const int r = tid >> 4;  // 16 threads per row
      const float4* src =
          (const float4*)(noise + ((size_t)(bn0 + r) * D_ + d) * TN_ + l0c);
      for (int c4 = (tid & 15); c4 < TILE_T / 4; c4 += 16)
        *(float4*)&Xs[r * PITCH + c4 * 4] = src[c4];
    }
    // hint-prefetch next channel's rows (global_prefetch_b8)
    if (d + 1 < D_ && tid < 16)
      __builtin_prefetch(noise + ((size_t)(bn0 + tid) * D_ + (d + 1)) * TN_ + l0c, 0, 0);
    __syncthreads();

    // ---- banded-Toeplitz GEMM: pe_q tile C[m=bn, n=l], contraction t ----
    // A[m,t] = Xs[m][lrel + t]          (16x4 frag: m=lane&15, k=kbase+vgpr)
    // B[t,n] = wl[16 + t - n]           (4x16 frag: n=lane&15, k=kbase+vgpr)
    v8f c = {0.f, 0.f, 0.f, 0.f, 0.f, 0.f, 0.f, 0.f};
    const int arow = col * PITCH + lrel + kbase;
    const int wbase = 16 + kbase - col;
#pragma unroll
    for (int t0 = 0; t0 < CONTR; t0 += 4) {
      v2f a, bf;
      a.x  = Xs[arow + t0];
      a.y  = Xs[arow + t0 + 1];
      bf.x = wl[wbase + t0];
      bf.y = wl[wbase + t0 + 1];
      c = __builtin_amdgcn_wmma_f32_16x16x4_f32(false, a, false, bf,
                                                (short)0, c, false, false);
    }

    // ---- fused reductions over d ----
    const float qv = qry[((size_t)b * D_ + d) * L_ + l_abs];
    const float kv = key[((size_t)b * D_ + d) * L_ + l_abs];
#pragma unroll
    for (int v = 0; v < 8; ++v) {
      const int row = v + hi * 8;  // C/D layout: row = v (+8 for upper lanes)
      qacc[v] += qv * c[v];
      kacc[v] += kv * Xs[row * PITCH + lrel + col + K_ / 2];
    }
  }

  // ---- write qhat / khat (coalesced per half-wave) ----
#pragma unroll
  for (int v = 0; v < 8; ++v) {
    const size_t bn = (size_t)(bn0 + v + hi * 8);
    outq[bn * L_ + l_abs] = qacc[v] * inv_scale;
    outk[bn * L_ + l_abs] = kacc[v] * inv_scale;
  }
}

extern "C" void kernel_launch(void* const* d_in, const int* in_sizes, int n_in,
                              void* d_out, int out_size, void* d_ws, size_t ws_size,
                              hipStream_t stream) {
  const float* qry   = (const float*)d_in[0];  // (B, D, L)
  const float* key   = (const float*)d_in[1];  // (B, D, L)
  const float* noise = (const float*)d_in[2];  // (B*NUM, D, L+2K)
  const float* cw    = (const float*)d_in[3];  // (D, 1, K)
  float* outq = (float*)d_out;                      // (B, NUM, L)
  float* outk = (float*)d_out + (size_t)B_ * NUM_ * L_;

  dim3 grid(L_ / CHUNK, (B_ * NUM_) / 16);  // 32 x 8
  convspe_wmma_kernel<<<grid, BLOCK, 0, stream>>>(qry, key, noise, cw, outq, outk);
}